// TileBasedRenderer_68410239090719
// MI455X (gfx1250) — compile-verified
//
#include <hip/hip_runtime.h>
#include <math.h>

typedef __attribute__((ext_vector_type(2))) float v2f;
typedef __attribute__((ext_vector_type(8))) float v8f;
typedef __attribute__((ext_vector_type(4))) int   v4i;

#define WIMG 256
#define HIMG 256

// ---------------------------------------------------------------------------
// CDNA5 helpers
// ---------------------------------------------------------------------------

// 16-byte async global->LDS copy (ASYNCcnt path); falls back to a plain copy.
// OFF is an immediate instruction offset applied to BOTH addresses (ISA 10.x).
template <int OFF>
__device__ __forceinline__ void async_cp16(void* lds, const void* g)
{
#if __has_builtin(__builtin_amdgcn_global_load_async_to_lds_b128)
    __builtin_amdgcn_global_load_async_to_lds_b128(
        (__attribute__((address_space(1))) v4i*)g,
        (__attribute__((address_space(3))) v4i*)lds,
        OFF, 0);
#else
    *(float4*)((char*)lds + OFF) = *(const float4*)((const char*)g + OFF);
#endif
}

__device__ __forceinline__ void async_wait_all()
{
#if __has_builtin(__builtin_amdgcn_s_wait_asynccnt)
    __builtin_amdgcn_s_wait_asynccnt(0);
#elif __has_builtin(__builtin_amdgcn_global_load_async_to_lds_b128)
    asm volatile("s_wait_asynccnt 0x0" ::: "memory");
#endif
}

// Swap values between lane p and lane p^16 of a wave32 (pure VALU on CDNA5).
__device__ __forceinline__ float half_swap(float v)
{
#if __has_builtin(__builtin_amdgcn_permlanex16)
    int iv = __float_as_int(v);
    int r  = __builtin_amdgcn_permlanex16(iv, iv, 0x76543210, 0xfedcba98, false, false);
    return __int_as_float(r);
#else
    return __shfl_xor(v, 16, 32);
#endif
}

// ---------------------------------------------------------------------------
// Kernel 1: per-gaussian preprocessing. 14 arrays of stride N in ws:
//  0..5 coefs, 6 u, 7 v, 8 radius, 9..11 rgb, 12 opacity(masked), 13 depth
// ---------------------------------------------------------------------------
__global__ void gs_preprocess(const float* __restrict__ pos,
                              const float* __restrict__ scl,
                              const float* __restrict__ rot,
                              const float* __restrict__ col,
                              const float* __restrict__ opa,
                              const float* __restrict__ vm,
                              float* __restrict__ ws, int N)
{
    int i = blockIdx.x * blockDim.x + threadIdx.x;
    if (i >= N) return;

    const float FX = 300.0f, FY = 300.0f;
    const float CXc = WIMG * 0.5f, CYc = HIMG * 0.5f;

    float p0 = pos[3*i+0], p1 = pos[3*i+1], p2 = pos[3*i+2];
    float pc[3];
    #pragma unroll
    for (int r = 0; r < 3; ++r)
        pc[r] = vm[r*4+0]*p0 + vm[r*4+1]*p1 + vm[r*4+2]*p2 + vm[r*4+3];
    float x = pc[0], y = pc[1], z = pc[2];
    float depth = -z;
    float sa = z + 1e-8f;
    float sgn = (sa > 0.f) ? 1.f : ((sa < 0.f) ? -1.f : 0.f);
    float zs = fmaxf(fabsf(z), 0.01f) * sgn;
    float z2 = zs * zs;

    float qw = rot[4*i+0], qx = rot[4*i+1], qy = rot[4*i+2], qz = rot[4*i+3];
    float qn = rsqrtf(qw*qw + qx*qx + qy*qy + qz*qz);
    qw *= qn; qx *= qn; qy *= qn; qz *= qn;
    float Rm[3][3] = {
        {1.f-2.f*(qy*qy+qz*qz), 2.f*(qx*qy-qw*qz),     2.f*(qx*qz+qw*qy)},
        {2.f*(qx*qy+qw*qz),     1.f-2.f*(qx*qx+qz*qz), 2.f*(qy*qz-qw*qx)},
        {2.f*(qx*qz-qw*qy),     2.f*(qy*qz+qw*qx),     1.f-2.f*(qx*qx+qy*qy)}};

    float Rc[3][3];
    #pragma unroll
    for (int r = 0; r < 3; ++r)
        #pragma unroll
        for (int cc = 0; cc < 3; ++cc)
            Rc[r][cc] = vm[r*4+0]*Rm[0][cc] + vm[r*4+1]*Rm[1][cc] + vm[r*4+2]*Rm[2][cc];

    float s0 = scl[3*i+0], s1 = scl[3*i+1], s2 = scl[3*i+2];
    float RS[3][3];
    #pragma unroll
    for (int r = 0; r < 3; ++r) {
        RS[r][0] = Rc[r][0]*s0; RS[r][1] = Rc[r][1]*s1; RS[r][2] = Rc[r][2]*s2;
    }
    float cov[3][3];
    #pragma unroll
    for (int r = 0; r < 3; ++r)
        #pragma unroll
        for (int cc = 0; cc < 3; ++cc)
            cov[r][cc] = RS[r][0]*RS[cc][0] + RS[r][1]*RS[cc][1] + RS[r][2]*RS[cc][2];

    float J00 = FX / (-zs), J02 = FX * x / z2;
    float J11 = FY /  zs,   J12 = FY * y / z2;
    float M00 = J00*cov[0][0] + J02*cov[2][0];
    float M01 = J00*cov[0][1] + J02*cov[2][1];
    float M02 = J00*cov[0][2] + J02*cov[2][2];
    float M10 = J11*cov[1][0] + J12*cov[2][0];
    float M11 = J11*cov[1][1] + J12*cov[2][1];
    float M12 = J11*cov[1][2] + J12*cov[2][2];
    float a = M00*J00 + M02*J02;
    float b = M01*J11 + M02*J12;
    float c = M10*J00 + M12*J02;
    float d = M11*J11 + M12*J12;

    float u = FX * x / (-zs) + CXc;
    float v = FY * y /  zs   + CYc;

    float tr = a + d;
    float det = fmaxf(a*d - b*c, 1e-6f);
    float disc = fmaxf(tr*tr - 4.f*det, 0.f);
    float lam = 0.5f * (tr + sqrtf(disc));
    float radius = fminf(3.f * sqrtf(fmaxf(lam, 1e-6f)), 64.f);

    bool vis = (depth > 0.01f) && (depth < 100.f) &&
               (u + radius > 0.f) && (u - radius < (float)WIMG) &&
               (v + radius > 0.f) && (v - radius < (float)HIMG);
    float o = vis ? opa[i] : 0.f;

    float ar = a + 1e-4f, dr = d + 1e-4f;
    float invdet = 1.f / (ar*dr - b*c);
    float ia  =  dr * invdet;
    float ib  = -b  * invdet;
    float ic  = -c  * invdet;
    float idd =  ar * invdet;
    float Bq = ib + ic;
    float cxx = -0.5f * ia;
    float cyy = -0.5f * idd;
    float cxy = -0.5f * Bq;
    float cx  = ia*u + 0.5f*Bq*v;
    float cy  = idd*v + 0.5f*Bq*u;
    float c1  = -0.5f * (ia*u*u + idd*v*v + Bq*u*v);

    ws[0*N+i]  = cxx;  ws[1*N+i]  = cyy;  ws[2*N+i]  = cxy;
    ws[3*N+i]  = cx;   ws[4*N+i]  = cy;   ws[5*N+i]  = c1;
    ws[6*N+i]  = u;    ws[7*N+i]  = v;    ws[8*N+i]  = radius;
    ws[9*N+i]  = col[3*i+0]; ws[10*N+i] = col[3*i+1]; ws[11*N+i] = col[3*i+2];
    ws[12*N+i] = o;    ws[13*N+i] = depth;
}

// ---------------------------------------------------------------------------
// Kernel 2: single-block bitonic argsort of depth (ascending), 4096 keys.
// ---------------------------------------------------------------------------
#define SORT_N 4096
__global__ __launch_bounds__(1024) void gs_sort(const float* __restrict__ depth,
                                                int* __restrict__ idx_out, int N)
{
    __shared__ float key[SORT_N];
    __shared__ int   sid[SORT_N];
    for (int i = threadIdx.x; i < SORT_N; i += blockDim.x) {
        key[i] = (i < N) ? depth[i] : 3.4e38f;
        sid[i] = i;
    }
    __syncthreads();
    for (int k = 2; k <= SORT_N; k <<= 1) {
        for (int j = k >> 1; j > 0; j >>= 1) {
            for (int i = threadIdx.x; i < SORT_N; i += blockDim.x) {
                int ij = i ^ j;
                if (ij > i) {
                    bool up = ((i & k) == 0);
                    float ki = key[i], kj = key[ij];
                    if ((ki > kj) == up) {
                        key[i] = kj; key[ij] = ki;
                        int t = sid[i]; sid[i] = sid[ij]; sid[ij] = t;
                    }
                }
            }
            __syncthreads();
        }
    }
    for (int i = threadIdx.x; i < N; i += blockDim.x) idx_out[i] = sid[i];
}

// ---------------------------------------------------------------------------
// Kernel 3: gather into depth-sorted order, packed for the render kernel:
//  Apk[g][8] = {cf0..cf5, 0, 0}   (WMMA A rows; zero pad feeds hi-half K)
//  Gpk[g][8] = {u, v, rad, op, r, g, b, 0}
// ---------------------------------------------------------------------------
__global__ void gs_gather(const float* __restrict__ un, const int* __restrict__ idx,
                          float* __restrict__ Apk, float* __restrict__ Gpk, int N)
{
    int i = blockIdx.x * blockDim.x + threadIdx.x;
    if (i >= N) return;
    int s = idx[i];
    Apk[i*8+0] = un[0*N+s]; Apk[i*8+1] = un[1*N+s];
    Apk[i*8+2] = un[2*N+s]; Apk[i*8+3] = un[3*N+s];
    Apk[i*8+4] = un[4*N+s]; Apk[i*8+5] = un[5*N+s];
    Apk[i*8+6] = 0.f;       Apk[i*8+7] = 0.f;
    Gpk[i*8+0] = un[6*N+s]; Gpk[i*8+1] = un[7*N+s];
    Gpk[i*8+2] = un[8*N+s]; Gpk[i*8+3] = un[12*N+s];
    Gpk[i*8+4] = un[9*N+s]; Gpk[i*8+5] = un[10*N+s];
    Gpk[i*8+6] = un[11*N+s]; Gpk[i*8+7] = 0.f;
}

// ---------------------------------------------------------------------------
// Kernel 4: render. Wave32 owns 32 consecutive pixels of one row.
// Double-buffered LDS chunks (256 gaussians) streamed via async global->LDS;
// per 16-gaussian batch: 4x V_WMMA_F32_16X16X4_F32 power tiles,
// v_permlanex16 half-tile exchange, strictly in-order blend.
// The asm barrier pins the 16 extracted powers so the compiler cannot
// rematerialize the WMMAs per-gaussian (observed in round 3).
// ---------------------------------------------------------------------------
__global__ __launch_bounds__(256) void gs_render(const float* __restrict__ Apk,
                                                 const float* __restrict__ Gpk,
                                                 float* __restrict__ out, int N)
{
    __shared__ __align__(16) float s_ac[2][256*8];   // 8 KB x2
    __shared__ __align__(16) float s_gp[2][256*8];   // 8 KB x2

    int tid  = threadIdx.x;
    int lane = tid & 31;
    int wgid = blockIdx.x * 8 + (tid >> 5);
    int y    = wgid >> 3;
    int x0   = (wgid & 7) * 32;
    int l15  = lane & 15;
    bool lo  = lane < 16;
    int aoff = lo ? 0 : 2;       // half-wave K-chunk selector (floats)

    float pyf = (float)y;
    float pxA = (float)(x0 + l15);
    float pxB = (float)(x0 + 16 + l15);
    float fA0 = pxA*pxA, fA1 = pyf*pyf, fA2 = pxA*pyf, fA3 = pxA, fA4 = pyf;
    float fB0 = pxB*pxB, fB2 = pxB*pyf, fB3 = pxB;

    v2f b1, b1b, b2, b2b;
    b1[0]  = lo ? fA0 : fA2;  b1[1]  = lo ? fA1 : fA3;
    b1b[0] = lo ? fA4 : 0.f;  b1b[1] = lo ? 1.f : 0.f;
    b2[0]  = lo ? fB0 : fB2;  b2[1]  = lo ? fA1 : fB3;
    b2b[0] = b1b[0];          b2b[1] = b1b[1];

    float pxO = (float)(x0 + lane);
    float accr = 0.f, accg = 0.f, accb = 0.f, acca = 0.f;

    int nchunks = (N + 255) >> 8;

    // issue chunk 0 into buffer 0
    {
        const char* gA = (const char*)Apk + (size_t)tid * 16;
        const char* gG = (const char*)Gpk + (size_t)tid * 16;
        char* lA = (char*)&s_ac[0][0] + tid * 16;
        char* lG = (char*)&s_gp[0][0] + tid * 16;
        async_cp16<0>(lA, gA);    async_cp16<4096>(lA, gA);
        async_cp16<0>(lG, gG);    async_cp16<4096>(lG, gG);
    }
    async_wait_all();
    __syncthreads();

    for (int ch = 0; ch < nchunks; ++ch) {
        int buf = ch & 1;
        if (ch + 1 < nchunks) {   // prefetch next chunk into the other buffer
            const char* gA = (const char*)Apk + (size_t)(ch + 1) * 8192 + tid * 16;
            const char* gG = (const char*)Gpk + (size_t)(ch + 1) * 8192 + tid * 16;
            char* lA = (char*)&s_ac[buf ^ 1][0] + tid * 16;
            char* lG = (char*)&s_gp[buf ^ 1][0] + tid * 16;
            async_cp16<0>(lA, gA);    async_cp16<4096>(lA, gA);
            async_cp16<0>(lG, gG);    async_cp16<4096>(lG, gG);
        }

        for (int b = 0; b < 16; ++b) {
            int gb = b * 16 + l15;
            const float* ac = &s_ac[buf][gb * 8];
            v2f a1 = *(const v2f*)(ac + aoff);        // one ds_load_b64
            v2f a2 = *(const v2f*)(ac + 4 + aoff);    // hi half reads the zero pad

            v8f zc = {0.f,0.f,0.f,0.f,0.f,0.f,0.f,0.f};
            v8f d1 = __builtin_amdgcn_wmma_f32_16x16x4_f32(false, a1, false, b1,  (short)0, zc, false, false);
            d1     = __builtin_amdgcn_wmma_f32_16x16x4_f32(false, a2, false, b1b, (short)0, d1, false, false);
            v8f d2 = __builtin_amdgcn_wmma_f32_16x16x4_f32(false, a1, false, b2,  (short)0, zc, false, false);
            d2     = __builtin_amdgcn_wmma_f32_16x16x4_f32(false, a2, false, b2b, (short)0, d2, false, false);

            float powLo[8], powHi[8];
            #pragma unroll
            for (int r = 0; r < 8; ++r) {
                float send = lo ? d2[r] : d1[r];
                float got  = half_swap(send);
                powLo[r] = lo ? d1[r] : got;
                powHi[r] = lo ? got : d2[r];
                // Optimization barrier: pin the extracted powers so the
                // compiler cannot rematerialize the WMMA tile per gaussian.
                asm volatile("" : "+v"(powLo[r]), "+v"(powHi[r]));
            }

            #pragma unroll
            for (int gg = 0; gg < 16; ++gg) {
                float p = (gg < 8) ? powLo[gg] : powHi[gg - 8];
                const float* gp = &s_gp[buf][(b * 16 + gg) * 8];
                float4 q0 = *(const float4*)gp;        // u, v, rad, op
                float4 q1 = *(const float4*)(gp + 4);  // r, g, b, pad
                int ins = (int)(pxO >= truncf(q0.x - q0.z)) &
                          (int)(pxO <= truncf(q0.x + q0.z)) &
                          (int)(pyf >= truncf(q0.y - q0.z)) &
                          (int)(pyf <= truncf(q0.y + q0.z));
                float alpha = ins ? q0.w * expf(p) : 0.f;
                float w = alpha * (1.f - acca);
                accr += w * q1.x;
                accg += w * q1.y;
                accb += w * q1.z;
                acca += w;
            }
        }
        async_wait_all();
        __syncthreads();
    }

    int pix = y * WIMG + x0 + lane;
    out[0 * WIMG * HIMG + pix] = accr;
    out[1 * WIMG * HIMG + pix] = accg;
    out[2 * WIMG * HIMG + pix] = accb;
}

// ---------------------------------------------------------------------------
extern "C" void kernel_launch(void* const* d_in, const int* in_sizes, int n_in,
                              void* d_out, int out_size, void* d_ws, size_t ws_size,
                              hipStream_t stream)
{
    const float* pos = (const float*)d_in[0];
    const float* scl = (const float*)d_in[1];
    const float* rot = (const float*)d_in[2];
    const float* col = (const float*)d_in[3];
    const float* opa = (const float*)d_in[4];
    const float* vm  = (const float*)d_in[5];
    float* out = (float*)d_out;
    int N = in_sizes[0] / 3;   // 4096

    float* wsf  = (float*)d_ws;
    float* un   = wsf;                          // 14 arrays of N
    float* Apk  = wsf + (size_t)14 * N;         // N*8
    float* Gpk  = wsf + (size_t)22 * N;         // N*8
    int*   sidx = (int*)(wsf + (size_t)30 * N);

    int blocks = (N + 255) / 256;
    gs_preprocess<<<blocks, 256, 0, stream>>>(pos, scl, rot, col, opa, vm, un, N);
    gs_sort<<<1, 1024, 0, stream>>>(un + (size_t)13 * N, sidx, N);
    gs_gather<<<blocks, 256, 0, stream>>>(un, sidx, Apk, Gpk, N);
    gs_render<<<WIMG * HIMG / (32 * 8), 256, 0, stream>>>(Apk, Gpk, out, N);
}